// ShortDeform_30090540875956
// MI455X (gfx1250) — compile-verified
//
#include <hip/hip_runtime.h>
#include <math.h>

// Shapes (fixed by the reference): B=2, obj=4, N=8, C=256, H=W=48
#define WDIM 48
#define HW   2304

typedef __attribute__((ext_vector_type(2))) float v2f;
typedef __attribute__((ext_vector_type(8))) float v8f;

__device__ __forceinline__ int iclamp(int v, int lo, int hi) {
    return v < lo ? lo : (v > hi ? hi : v);
}

// D = A(16x4 f32) * B(4x16 f32) + C(16x16 f32), exact fp32 matrix op on CDNA5.
__device__ __forceinline__ v8f wmma_k4(v2f a, v2f b, v8f c) {
    return __builtin_amdgcn_wmma_f32_16x16x4_f32(
        /*neg_a=*/false, a, /*neg_b=*/false, b,
        /*c_mod=*/(short)0, c, /*reuse_a=*/false, /*reuse_b=*/false);
}

// ---------------------------------------------------------------------------
// Weight repack: w[O][C][9] -> wt[t][c][opad]  (zero-fill o >= outC).
// Makes B-fragment loads lane-contiguous and guard-free in the GEMM loops.
// ---------------------------------------------------------------------------
__global__ void repack_w(const float* __restrict__ w, float* __restrict__ wt,
                         int inC, int outC, int outCpad)
{
    const int idx   = blockIdx.x * blockDim.x + threadIdx.x;
    const int total = 9 * inC * outCpad;
    if (idx >= total) return;
    const int o = idx % outCpad;
    const int c = (idx / outCpad) % inC;
    const int t = idx / (outCpad * inC);
    wt[idx] = (o < outC) ? w[((size_t)o * inC + c) * 9 + t] : 0.f;
}

// ---------------------------------------------------------------------------
// K-range inner loop: channels [0,nc) of one input tensor, one 3x3 tap.
//   inp : plane of this range's channel 0 (for this batch), offset by spc
//   wq  : wt[t] + cbeg*outCpad + (n0 + l)   (this lane's B column)
// NT accumulator tiles of 16 out-channels each; A amortized over NT WMMAs.
// ---------------------------------------------------------------------------
template<int NT>
__device__ __forceinline__ void conv_krange(const float* __restrict__ inp, int nc,
                                            const float* __restrict__ wq, int outCpad,
                                            float amask, int spc, int half,
                                            v8f acc[NT])
{
    const float* __restrict__ p = inp + (size_t)(2 * half) * HW + spc;
    const float* __restrict__ q = wq + (size_t)(2 * half) * outCpad;
#pragma unroll 2
    for (int c0 = 0; c0 < nc; c0 += 4) {
        v2f A;
        A.x = amask * p[0];
        A.y = amask * p[HW];
        __builtin_prefetch(p + (size_t)8 * HW, 0, 1);
#pragma unroll
        for (int j = 0; j < NT; ++j) {
            v2f Bv;
            Bv.x = q[j * 16];
            Bv.y = q[outCpad + j * 16];
            acc[j] = wmma_k4(A, Bv, acc[j]);
        }
        p += (size_t)4 * HW;
        q += (size_t)4 * outCpad;
    }
}

// ---------------------------------------------------------------------------
// 3x3 SAME conv as implicit GEMM. One wave computes 16 pixels x (16*NT) outC.
// Channel-split inputs handle concat(query,key) without per-iter selects.
// ---------------------------------------------------------------------------
template<int NT>
__global__ __launch_bounds__(32)
void conv3x3_wmma(const float* __restrict__ in0, const float* __restrict__ in1,
                  int splitC, int inC, int outC, int outCpad,
                  const float* __restrict__ wt, const float* __restrict__ bias,
                  float* __restrict__ out)
{
    const int lane = threadIdx.x;
    const int half = lane >> 4;          // K-half: lanes 0-15 -> K 0,1 ; 16-31 -> K 2,3
    const int l    = lane & 15;
    const int m0   = blockIdx.x * 16;
    const int n0   = blockIdx.y * 16 * NT;
    const int b    = blockIdx.z;
    const int c1   = inC - splitC;

    const int pix = m0 + l;
    const int py  = pix / WDIM;
    const int px  = pix - py * WDIM;

    v8f acc[NT];
#pragma unroll
    for (int j = 0; j < NT; ++j) acc[j] = (v8f){0.f,0.f,0.f,0.f,0.f,0.f,0.f,0.f};

    const float* __restrict__ p0 = in0 + (size_t)b * splitC * HW;
    const float* __restrict__ p1 = in1 + (size_t)b * c1 * HW;

    for (int t = 0; t < 9; ++t) {
        const int dy = t / 3 - 1;
        const int dx = t - (t / 3) * 3 - 1;
        const int yy = py + dy, xx = px + dx;
        // branch-free zero padding: clamped address + multiplicative mask
        const float amask = ((yy >= 0) & (yy < WDIM) & (xx >= 0) & (xx < WDIM)) ? 1.f : 0.f;
        const int spc = iclamp(yy, 0, WDIM - 1) * WDIM + iclamp(xx, 0, WDIM - 1);

        const float* __restrict__ wtt = wt + (size_t)t * inC * outCpad;
        conv_krange<NT>(p0, splitC, wtt + n0 + l, outCpad, amask, spc, half, acc);
        if (c1 > 0)
            conv_krange<NT>(p1, c1, wtt + (size_t)splitC * outCpad + n0 + l,
                            outCpad, amask, spc, half, acc);
    }

#pragma unroll
    for (int j = 0; j < NT; ++j) {
        const int oc = n0 + j * 16 + l;
        if (oc < outC) {
            const float bz = bias[oc];
            float* __restrict__ o = out + ((size_t)b * outC + oc) * HW + m0 + 8 * half;
#pragma unroll
            for (int i = 0; i < 8; ++i)          // D: VGPR i -> M = i + 8*half
                o[i] = acc[j][i] + bz;
        }
    }
}

// ---------------------------------------------------------------------------
// Deformable 3x3 conv, fused bilinear gather + GEMM + bias + query + transpose.
// One wave: 16 pixels x 64 outC (4 accumulators) -> each 8-gather bilinear
// A-fragment feeds 4 WMMAs.
// value: [8,256,2304]  off: [2,18,2304]  wt: [9][256][256] (t,c,o repacked)
// ---------------------------------------------------------------------------
__global__ __launch_bounds__(32)
void deform_wmma(const float* __restrict__ value, const float* __restrict__ off,
                 const float* __restrict__ wt, const float* __restrict__ bias,
                 const float* __restrict__ query, float* __restrict__ out)
{
    const int lane = threadIdx.x;
    const int half = lane >> 4;
    const int l    = lane & 15;
    const int m0   = blockIdx.x * 16;
    const int n0   = blockIdx.y * 64;
    const int n    = blockIdx.z;          // 0..7
    const int b    = n >> 2;              // n = b*obj + j

    const int pix = m0 + l;
    const int py  = pix / WDIM;
    const int px  = pix - py * WDIM;

    const float* __restrict__ vbase = value + (size_t)n * 256 * HW;

    v8f acc[4];
#pragma unroll
    for (int j = 0; j < 4; ++j) acc[j] = (v8f){0.f,0.f,0.f,0.f,0.f,0.f,0.f,0.f};

    for (int t = 0; t < 9; ++t) {
        // sampling coords: off + integer grid + kernel tap (matches reference)
        const float dy  = off[((size_t)b * 18 + 2 * t) * HW + pix];
        const float dx  = off[((size_t)b * 18 + 2 * t + 1) * HW + pix];
        const float pyf = dy + (float)(py + t / 3 - 1);
        const float pxf = dx + (float)(px + (t % 3) - 1);
        const float y0f = floorf(pyf), x0f = floorf(pxf);
        const float fy  = pyf - y0f,   fx  = pxf - x0f;
        const int   y0  = (int)y0f,    x0  = (int)x0f;

        const bool vy0 = (y0 >= 0) & (y0 < WDIM);
        const bool vy1 = (y0 + 1 >= 0) & (y0 + 1 < WDIM);
        const bool vx0 = (x0 >= 0) & (x0 < WDIM);
        const bool vx1 = (x0 + 1 >= 0) & (x0 + 1 < WDIM);

        // zero-pad validity folded into bilinear weights (branch-free)
        const float w00 = (vy0 & vx0) ? (1.f - fy) * (1.f - fx) : 0.f;
        const float w01 = (vy0 & vx1) ? (1.f - fy) * fx         : 0.f;
        const float w10 = (vy1 & vx0) ? fy * (1.f - fx)         : 0.f;
        const float w11 = (vy1 & vx1) ? fy * fx                 : 0.f;

        const int yc0 = iclamp(y0, 0, WDIM - 1), yc1 = iclamp(y0 + 1, 0, WDIM - 1);
        const int xc0 = iclamp(x0, 0, WDIM - 1), xc1 = iclamp(x0 + 1, 0, WDIM - 1);
        const int i00 = yc0 * WDIM + xc0, i01 = yc0 * WDIM + xc1;
        const int i10 = yc1 * WDIM + xc0, i11 = yc1 * WDIM + xc1;

        const float* __restrict__ vp = vbase + (size_t)(2 * half) * HW;
        const float* __restrict__ q  = wt + ((size_t)t * 256 + 2 * half) * 256 + n0 + l;

#pragma unroll 2
        for (int c0 = 0; c0 < 256; c0 += 4) {
            const float* __restrict__ vq = vp + HW;
            v2f A;
            A.x = w00 * vp[i00] + w01 * vp[i01] + w10 * vp[i10] + w11 * vp[i11];
            A.y = w00 * vq[i00] + w01 * vq[i01] + w10 * vq[i10] + w11 * vq[i11];
            __builtin_prefetch(vp + (size_t)8 * HW + i00, 0, 1);
#pragma unroll
            for (int j = 0; j < 4; ++j) {
                v2f Bv;
                Bv.x = q[j * 16];
                Bv.y = q[256 + j * 16];
                acc[j] = wmma_k4(A, Bv, acc[j]);
            }
            vp += (size_t)4 * HW;
            q  += (size_t)4 * 256;
        }
    }

    // fused epilogue: out[pix, n, oc] = deform + bias + query  (does the transpose)
#pragma unroll
    for (int j = 0; j < 4; ++j) {
        const int oc = n0 + j * 16 + l;
        const float bz = bias[oc];
        const float* __restrict__ qv = query + ((size_t)b * 256 + oc) * HW;
#pragma unroll
        for (int i = 0; i < 8; ++i) {
            const int pp = m0 + i + 8 * half;     // D row M = i + 8*half
            out[(size_t)pp * (8 * 256) + (size_t)n * 256 + oc] = acc[j][i] + bz + qv[pp];
        }
    }
}

// ---------------------------------------------------------------------------
extern "C" void kernel_launch(void* const* d_in, const int* in_sizes, int n_in,
                              void* d_out, int out_size, void* d_ws, size_t ws_size,
                              hipStream_t stream) {
    const float* query = (const float*)d_in[0];   // [2,256,48,48]
    const float* key   = (const float*)d_in[1];   // [2,256,48,48]
    const float* value = (const float*)d_in[2];   // [2,4,256,48,48]
    const float* w1    = (const float*)d_in[3];   // [256,512,3,3]
    const float* b1    = (const float*)d_in[4];   // [256]
    const float* w2    = (const float*)d_in[5];   // [18,256,3,3]
    const float* b2    = (const float*)d_in[6];   // [18]
    const float* wd    = (const float*)d_in[7];   // [256,256,3,3]
    const float* bd    = (const float*)d_in[8];   // [256]

    // workspace layout (floats)
    float* feat = (float*)d_ws;                          // 2*256*2304   = 1,179,648
    float* off  = feat + (size_t)2 * 256 * HW;           // 2*18*2304    =    82,944
    float* wt1  = off  + (size_t)2 * 18 * HW;            // 9*512*256    = 1,179,648
    float* wt2  = wt1  + (size_t)9 * 512 * 256;          // 9*256*32     =    73,728
    float* wtd  = wt2  + (size_t)9 * 256 * 32;           // 9*256*256    =   589,824
                                                         // total ~12.4 MB

    const dim3 blk(32);

    // weight repacks (tiny, fully parallel)
    {
        int tot1 = 9 * 512 * 256;
        repack_w<<<(tot1 + 255) / 256, 256, 0, stream>>>(w1, wt1, 512, 256, 256);
        int tot2 = 9 * 256 * 32;
        repack_w<<<(tot2 + 255) / 256, 256, 0, stream>>>(w2, wt2, 256, 18, 32);
        int totd = 9 * 256 * 256;
        repack_w<<<(totd + 255) / 256, 256, 0, stream>>>(wd, wtd, 256, 256, 256);
    }

    // conv1: concat(query,key) [B,512] -> feat [B,256]   (NT=4 -> 64 outC/wave)
    conv3x3_wmma<4><<<dim3(HW / 16, 4, 2), blk, 0, stream>>>(
        query, key, /*splitC=*/256, /*inC=*/512, /*outC=*/256, /*outCpad=*/256,
        wt1, b1, feat);

    // conv2: feat [B,256] -> off [B,18]  (obj-broadcast exploited; NT=2 covers 32)
    conv3x3_wmma<2><<<dim3(HW / 16, 1, 2), blk, 0, stream>>>(
        feat, feat, /*splitC=*/256, /*inC=*/256, /*outC=*/18, /*outCpad=*/32,
        wt2, b2, off);

    // deformable conv + fused epilogue -> out [2304, 8, 256]
    deform_wmma<<<dim3(HW / 16, 4, 8), blk, 0, stream>>>(
        value, off, wtd, bd, query, (float*)d_out);
}